// PPOGNNPolicy_30949534335114
// MI455X (gfx1250) — compile-verified
//
#include <hip/hip_runtime.h>

typedef __attribute__((ext_vector_type(2))) float v2f;
typedef __attribute__((ext_vector_type(8))) float v8f;
typedef __attribute__((ext_vector_type(4))) int   v4i;

#define AS1 __attribute__((address_space(1)))
#define AS3 __attribute__((address_space(3)))

#define HID   256
#define FNODE 32
#define FEDGE 8
#define GDIM  16
#define NACT  64
#define NB    64

#if __has_builtin(__builtin_amdgcn_global_load_async_to_lds_b128)
#define HAVE_ASYNC_LDS 1
#else
#define HAVE_ASYNC_LDS 0
#endif

__device__ __forceinline__ void wait_async_zero() {
#if __has_builtin(__builtin_amdgcn_s_wait_asynccnt)
  __builtin_amdgcn_s_wait_asynccnt(0);
#else
  asm volatile("s_wait_asynccnt 0x0" ::: "memory");
#endif
}

// ---------------------------------------------------------------- zero fill
__global__ void zero_kernel(float* __restrict__ p, long n) {
  long i = (long)blockIdx.x * blockDim.x + threadIdx.x;
  long stride = (long)gridDim.x * blockDim.x;
  for (; i < n; i += stride) p[i] = 0.0f;
}

// ------------------------------------------------- layer 0 edge message (d=32)
// one wave per edge; lane = output channel. We0 (8x32) staged in LDS.
__global__ void edge_msg_l0(const float* __restrict__ x, const float* __restrict__ ea,
                            const float* __restrict__ We0, const int* __restrict__ ei,
                            float* __restrict__ agg, int E_) {
  __shared__ float sW[FEDGE * FNODE];
  for (int i = threadIdx.x; i < FEDGE * FNODE; i += blockDim.x) sW[i] = We0[i];
  __syncthreads();
  int lane = threadIdx.x & 31;
  int wv = threadIdx.x >> 5;
  int e = blockIdx.x * (blockDim.x >> 5) + wv;
  if (e >= E_) return;
  int src = ei[e];
  int dst = ei[E_ + e];
  float acc = x[src * FNODE + lane];
#pragma unroll
  for (int k = 0; k < FEDGE; k++) {
    float ev = ea[e * FEDGE + k];      // broadcast load, cached
    acc += ev * sW[k * FNODE + lane];
  }
  acc = acc > 0.0f ? acc : 0.0f;
  atomicAdd(&agg[dst * FNODE + lane], acc);
}

// ------------------------------------------------- hidden edge message (d=256)
// one wave per edge; lane covers 8 strided channels. We (8x256=8KB) in LDS.
__global__ void edge_msg_h(const float* __restrict__ h, const float* __restrict__ ea,
                           const float* __restrict__ We, const int* __restrict__ ei,
                           float* __restrict__ agg, int E_) {
  __shared__ float sW[FEDGE * HID];
  for (int i = threadIdx.x; i < FEDGE * HID; i += blockDim.x) sW[i] = We[i];
  __syncthreads();
  int lane = threadIdx.x & 31;
  int wv = threadIdx.x >> 5;
  int e = blockIdx.x * (blockDim.x >> 5) + wv;
  if (e >= E_) return;
  int src = ei[e];
  int dst = ei[E_ + e];
  float ev[FEDGE];
#pragma unroll
  for (int k = 0; k < FEDGE; k++) ev[k] = ea[e * FEDGE + k];
  const float* __restrict__ hrow = h + (long)src * HID;
  float* __restrict__ arow = agg + (long)dst * HID;
#pragma unroll
  for (int i = 0; i < HID / 32; i++) {
    int c = lane + 32 * i;
    float acc = hrow[c];
#pragma unroll
    for (int k = 0; k < FEDGE; k++) acc += ev[k] * sW[k * HID + c];
    acc = acc > 0.0f ? acc : 0.0f;
    atomicAdd(&arow[c], acc);
  }
}

// --------------------------------- node update: relu((hin+agg) @ W + b) -> hout
// W is KxHID row-major. Workgroup (256 thr = 8 waves) computes 16 rows x 256 cols.
// A panel (16xK, contiguous in memory) DMA'd into LDS with async-to-LDS loads
// (ASYNCcnt), residual add fused in LDS; each wave owns 32 output columns and
// accumulates with V_WMMA_F32_16X16X4_F32 over K in steps of 4.
template <int K>
__global__ void node_gemm(const float* __restrict__ hin, const float* __restrict__ agg,
                          const float* __restrict__ W, const float* __restrict__ bias,
                          float* __restrict__ hout, int N_) {
  __shared__ float sA[16 * K];
  __shared__ float sG[16 * K];
  int rowBase = blockIdx.x * 16;

  if (rowBase + 16 <= N_) {
#if HAVE_ASYNC_LDS
    // whole 16xK panel is one contiguous 64*K-byte span: DMA it straight to LDS
    const float* hp = hin + (long)rowBase * K;
    const float* ap = agg + (long)rowBase * K;
    for (int c = threadIdx.x; c < 4 * K; c += blockDim.x) {  // b128 chunks
      __builtin_amdgcn_global_load_async_to_lds_b128(
          (AS1 v4i*)(hp + 4 * c), (AS3 v4i*)(&sA[4 * c]), 0, 0);
      __builtin_amdgcn_global_load_async_to_lds_b128(
          (AS1 v4i*)(ap + 4 * c), (AS3 v4i*)(&sG[4 * c]), 0, 0);
    }
    wait_async_zero();
#else
    for (int i = threadIdx.x; i < 16 * K; i += blockDim.x) {
      sA[i] = hin[(long)rowBase * K + i];
      sG[i] = agg[(long)rowBase * K + i];
    }
#endif
  } else {
    // partial tail block: per-element guarded loads, zero padding
    for (int i = threadIdx.x; i < 16 * K; i += blockDim.x) {
      int r = i / K, k = i % K;
      int row = rowBase + r;
      float hv = 0.0f, gv = 0.0f;
      if (row < N_) {
        hv = hin[(long)row * K + k];
        gv = agg[(long)row * K + k];
      }
      sA[i] = hv;
      sG[i] = gv;
    }
  }
  __syncthreads();
  // fused residual add: sA = hin + agg
  for (int i = threadIdx.x; i < 16 * K; i += blockDim.x) sA[i] += sG[i];
  __syncthreads();

  int lane = threadIdx.x & 31;
  int wave = threadIdx.x >> 5;       // 8 waves -> 8 * 32 = 256 columns
  int colBase = wave * 32;
  int l16 = lane & 15;
  int hi = lane >> 4;                // 0: K+{0,1} / M 0..7 ; 1: K+{2,3} / M 8..15

  v8f acc0 = {};
  v8f acc1 = {};
  for (int k0 = 0; k0 < K; k0 += 4) {
    int ka = k0 + 2 * hi;
    v2f a;                            // A 16x4: lanes0-15 K={0,1}, lanes16-31 K={2,3}
    a.x = sA[l16 * K + ka];
    a.y = sA[l16 * K + ka + 1];
    v2f b0, b1;                       // B 4x16: same K split, N = l16
    b0.x = W[(long)ka * HID + colBase + l16];
    b0.y = W[(long)(ka + 1) * HID + colBase + l16];
    b1.x = W[(long)ka * HID + colBase + 16 + l16];
    b1.y = W[(long)(ka + 1) * HID + colBase + 16 + l16];
    acc0 = __builtin_amdgcn_wmma_f32_16x16x4_f32(false, a, false, b0, (short)0, acc0,
                                                 false, false);
    acc1 = __builtin_amdgcn_wmma_f32_16x16x4_f32(false, a, false, b1, (short)0, acc1,
                                                 false, false);
  }

  // D layout: VGPR v -> M = v (lanes 0-15) / v+8 (lanes 16-31), N = lane%16
#pragma unroll
  for (int v = 0; v < 8; v++) {
    int row = rowBase + v + 8 * hi;
    if (row < N_) {
      int c0 = colBase + l16;
      float x0 = acc0[v] + bias[c0];
      hout[(long)row * HID + c0] = x0 > 0.0f ? x0 : 0.0f;
      int c1 = colBase + 16 + l16;
      float x1 = acc1[v] + bias[c1];
      hout[(long)row * HID + c1] = x1 > 0.0f ? x1 : 0.0f;
    }
  }
}

// ---------------------------------------------------------------- mean pool
__global__ void pool_kernel(const float* __restrict__ h, const int* __restrict__ batch,
                            float* __restrict__ pooled, float* __restrict__ counts, int N_) {
  long idx = (long)blockIdx.x * blockDim.x + threadIdx.x;
  int n = (int)(idx >> 8);
  int c = (int)(idx & 255);
  if (n < N_) {
    int g = batch[n];
    atomicAdd(&pooled[g * HID + c], h[(long)n * HID + c]);
    if (c == 0) atomicAdd(&counts[g], 1.0f);
  }
}

// --------------------------------------------------------------- policy head
__global__ void head_kernel(const float* __restrict__ pooled, const float* __restrict__ counts,
                            const float* __restrict__ gf, const float* __restrict__ Wp,
                            const float* __restrict__ bp, const float* __restrict__ Wv,
                            const float* __restrict__ bv, float* __restrict__ out) {
  __shared__ float f[HID + GDIM];
  int b = blockIdx.x;
  float cnt = counts[b];
  cnt = cnt > 1.0f ? cnt : 1.0f;
  for (int i = threadIdx.x; i < HID; i += blockDim.x) f[i] = pooled[b * HID + i] / cnt;
  for (int i = threadIdx.x; i < GDIM; i += blockDim.x) f[HID + i] = gf[b * GDIM + i];
  __syncthreads();
  int a = threadIdx.x;
  if (a < NACT) {
    float acc = bp[a];
    for (int k = 0; k < HID + GDIM; k++) acc += f[k] * Wp[k * NACT + a];
    out[b * NACT + a] = acc;
  } else if (a == NACT) {
    float acc = bv[0];
    for (int k = 0; k < HID + GDIM; k++) acc += f[k] * Wv[k];
    out[NB * NACT + b] = acc;
  }
}

// ------------------------------------------------------------------ launcher
extern "C" void kernel_launch(void* const* d_in, const int* in_sizes, int n_in,
                              void* d_out, int out_size, void* d_ws, size_t ws_size,
                              hipStream_t stream) {
  const float* x   = (const float*)d_in[0];
  const float* ea  = (const float*)d_in[1];
  const float* gf  = (const float*)d_in[2];
  const float* We0 = (const float*)d_in[3];
  const float* W0  = (const float*)d_in[4];
  const float* b0  = (const float*)d_in[5];
  const float* We  = (const float*)d_in[6];
  const float* W   = (const float*)d_in[7];
  const float* bb  = (const float*)d_in[8];
  const float* Wp  = (const float*)d_in[9];
  const float* bp  = (const float*)d_in[10];
  const float* Wv  = (const float*)d_in[11];
  const float* bv  = (const float*)d_in[12];
  const int*   ei  = (const int*)d_in[13];
  const int* batch = (const int*)d_in[14];

  int Nn = in_sizes[0] / FNODE;
  int Ee = in_sizes[1] / FEDGE;

  float* ws = (float*)d_ws;
  float* agg    = ws;                          // N*HID (layer0 uses first N*32)
  float* hA     = agg + (size_t)Nn * HID;      // N*HID
  float* hB     = hA + (size_t)Nn * HID;       // N*HID
  float* pooled = hB + (size_t)Nn * HID;       // NB*HID
  float* counts = pooled + NB * HID;           // NB

  dim3 blk(256);
  int ewaves = 256 / 32;
  int eblocks = (Ee + ewaves - 1) / ewaves;
  int nblocks = (Nn + 15) / 16;

  // layer 0: F_NODE=32 -> HID=256
  zero_kernel<<<2048, blk, 0, stream>>>(agg, (long)Nn * FNODE);
  edge_msg_l0<<<eblocks, blk, 0, stream>>>(x, ea, We0, ei, agg, Ee);
  node_gemm<FNODE><<<nblocks, blk, 0, stream>>>(x, agg, W0, b0, hA, Nn);

  float* cur = hA;
  float* nxt = hB;
  for (int l = 0; l < 3; l++) {
    zero_kernel<<<4096, blk, 0, stream>>>(agg, (long)Nn * HID);
    edge_msg_h<<<eblocks, blk, 0, stream>>>(cur, ea, We + (size_t)l * FEDGE * HID, ei, agg, Ee);
    node_gemm<HID><<<nblocks, blk, 0, stream>>>(cur, agg, W + (size_t)l * HID * HID,
                                                bb + l * HID, nxt, Nn);
    float* t = cur; cur = nxt; nxt = t;
  }

  zero_kernel<<<64, blk, 0, stream>>>(pooled, (long)(NB * HID + NB));
  pool_kernel<<<Nn, blk, 0, stream>>>(cur, batch, pooled, counts, Nn);
  head_kernel<<<NB, 128, 0, stream>>>(pooled, counts, gf, Wp, bp, Wv, bv, (float*)d_out);
}